// FrozenBNBEmbedding_8392366096544
// MI455X (gfx1250) — compile-verified
//
#include <hip/hip_runtime.h>
#include <cstdint>

#define THREADS 256
#define DIM     4096   // GPT-J hidden dim == bitsandbytes BLOCK, so absmax is per-row

// AMDGPU AS1 (global, prints as __device__) and AS3 (LDS, prints as __shared__)
typedef __attribute__((address_space(1))) int gint_t;
typedef __attribute__((address_space(3))) int lint_t;

// One workgroup (8 wave32s) dequantizes one token row of 4096 elements.
// Memory-bound: 16KB int8-codes(int32) in + 16KB fp32 out per token.
__global__ __launch_bounds__(THREADS) void dequant_embed_kernel(
    const int*   __restrict__ tok,     // [n_tok] token ids
    const int*   __restrict__ weight,  // [V, DIM] int8 codes stored as int32
    const float* __restrict__ absmax,  // [V] per-row (per-4096-block) scale
    const float* __restrict__ code,    // [256] dequant LUT
    float*       __restrict__ out)     // [n_tok, DIM]
{
    __shared__ float lut[256];
    const int tid = threadIdx.x;

    // --- Stage the 1KB LUT into LDS via the CDNA5 async global->LDS path ---
#if __has_builtin(__builtin_amdgcn_global_load_async_to_lds_b32)
    {
        gint_t* gsrc = (gint_t*)(uintptr_t)(code + tid);               // AS1 ptr
        lint_t* ldst = (lint_t*)(uint32_t)(uintptr_t)(&lut[tid]);      // AS3 ptr (low 32 bits = LDS offset)
        __builtin_amdgcn_global_load_async_to_lds_b32(gsrc, ldst, 0, 0);
        asm volatile("s_wait_asynccnt 0" ::: "memory");
    }
#else
    lut[tid] = code[tid];
#endif

    // Scalar (uniform per block): token id and its per-row scale.
    const int   token = tok[blockIdx.x];
    const float scale = absmax[token];
    const int*  __restrict__ wrow = weight + (size_t)token      * DIM;
    float*      __restrict__ orow = out    + (size_t)blockIdx.x * DIM;

    // Warm the gathered row while we wait on the LUT barrier.
    __builtin_prefetch(wrow + tid * 4, 0, 0);

    __syncthreads();

    // 4096 elems / 256 threads = 16 per thread, as 4 x b128 load/store pairs.
#pragma unroll
    for (int c = 0; c < DIM / (THREADS * 4); ++c) {
        const int base = c * (THREADS * 4) + tid * 4;
        const int4 q = *(const int4*)(wrow + base);
        float4 r;
        r.x = lut[q.x & 255] * scale;
        r.y = lut[q.y & 255] * scale;
        r.z = lut[q.z & 255] * scale;
        r.w = lut[q.w & 255] * scale;
        *(float4*)(orow + base) = r;
    }
}

extern "C" void kernel_launch(void* const* d_in, const int* in_sizes, int n_in,
                              void* d_out, int out_size, void* d_ws, size_t ws_size,
                              hipStream_t stream) {
    const int*   tok    = (const int*)  d_in[0];  // [4*2048] token ids
    const int*   weight = (const int*)  d_in[1];  // [50400*4096] codes
    const float* absmax = (const float*)d_in[2];  // [50400] scales
    const float* code   = (const float*)d_in[3];  // [256] LUT
    float*       out    = (float*)      d_out;    // [4*2048*4096] fp32

    const int n_tok = in_sizes[0];                // 8192
    dequant_embed_kernel<<<n_tok, THREADS, 0, stream>>>(tok, weight, absmax, code, out);
}